// GeCoLoss_58136677319223
// MI455X (gfx1250) — compile-verified
//
#include <hip/hip_runtime.h>
#include <hip/hip_bf16.h>
#include <math.h>

typedef float v2f __attribute__((ext_vector_type(2)));
typedef float v8f __attribute__((ext_vector_type(8)));
typedef unsigned int u32x4 __attribute__((ext_vector_type(4)));
typedef int i32x4 __attribute__((ext_vector_type(4)));
typedef int i32x8 __attribute__((ext_vector_type(8)));

#define EPSV 1e-6f
#define INV_T (1.0f / 0.07f)
#define NEG_BIG (-3.402823466e38f)

// GEMM LDS tiling: K processed in two 64-wide slabs; TDM pads +2 dwords per
// 64-dword row -> LDS row stride 66 dwords (8B aligned, bank-conflict-light).
#define QROWS 128
#define PROWS 64
#define LROW  66

// ---------------------------------------------------------------------------
// Tensor Data Mover: async 2D tile load global->LDS (D# per CDNA5 ISA ch.8).
// tile_cols elements/row (data_size=4B), nrows rows, row_stride elements,
// LDS padding: +2 dwords after every 64 dwords (interval code 5, amount code 1).
// ---------------------------------------------------------------------------
__device__ __forceinline__ void tdm_load_tile(unsigned lds_byte_off,
                                              const float* gptr,
                                              unsigned tile_cols,
                                              unsigned nrows,
                                              unsigned row_stride) {
  unsigned long long g = (unsigned long long)(size_t)(const void*)gptr;
  u32x4 g0;
  g0.x = 1u;                                                   // count=1 (valid user D#)
  g0.y = lds_byte_off;                                         // lds_addr
  g0.z = (unsigned)g;                                          // global_addr[31:0]
  g0.w = (unsigned)((g >> 32) & 0x01FFFFFFu) | (2u << 30);     // addr[56:32] | type=2
  i32x8 g1;
  g1[0] = (int)((2u << 16) | (1u << 20) | (5u << 22) | (1u << 25)); // data_size=4B, pad_en, ivl=64dw, amt=2dw
  g1[1] = (int)((tile_cols & 0xFFFFu) << 16);                  // tensor_dim0[15:0]
  g1[2] = (int)(((tile_cols >> 16) & 0xFFFFu) |
                ((nrows & 0xFFFFu) << 16));                    // tensor_dim0[31:16] | tensor_dim1[15:0]
  g1[3] = (int)(((nrows >> 16) & 0xFFFFu) |
                ((tile_cols & 0xFFFFu) << 16));                // tensor_dim1[31:16] | tile_dim0
  g1[4] = (int)(nrows & 0xFFFFu);                              // tile_dim1 (tile_dim2 = 0)
  g1[5] = (int)row_stride;                                     // tensor_dim0_stride[31:0]
  g1[6] = 0;                                                   // stride0[47:32] | stride1[15:0]
  g1[7] = 0;
  i32x4 z4 = {0, 0, 0, 0};
#if defined(__clang_major__) && (__clang_major__ >= 23)
  i32x8 z8 = {0, 0, 0, 0, 0, 0, 0, 0};
  __builtin_amdgcn_tensor_load_to_lds(g0, g1, z4, z4, z8, 0);
#else
  __builtin_amdgcn_tensor_load_to_lds(g0, g1, z4, z4, 0);
#endif
}

// ---------------------------------------------------------------------------
// K0: zero the accumulator region (per-batch loss sums + contrastive sums)
// ---------------------------------------------------------------------------
__global__ void k_zero(float* __restrict__ p, int n) {
  int i = blockIdx.x * blockDim.x + threadIdx.x;
  if (i < n) p[i] = 0.0f;
}

// ---------------------------------------------------------------------------
// K1: sim[b,l,s] = dot(q_geo[b,l,:], geo_pos[b,s,:])
// Block: 128(M) x 64(N) tile, 8 waves (one 16-row strip each).
// TDM stages Q(128x64 K-slab) and P(64x64 K-slab) into LDS; waves read
// fragments via ds_load_b64 and issue v_wmma_f32_16x16x4_f32.
// A frag: lane holds Qs[wave*16 + lane%16][ko..ko+1], ko = k + 2*(lane>=16)
// B frag: lane holds Ps[j*16 + lane%16][ko..ko+1]
// C/D: vgpr i -> row m0+wave*16+i+8*(lane>=16), col n0 + j*16 + lane%16
// ---------------------------------------------------------------------------
__global__ __launch_bounds__(256) void k_sim_gemm(const float* __restrict__ Q,
                                                  const float* __restrict__ P,
                                                  float* __restrict__ sim) {
  __shared__ __align__(16) float lds[QROWS * LROW + PROWS * LROW];  // 50688 B
  const int b    = blockIdx.z;
  const int wave = threadIdx.x >> 5;
  const int lane = threadIdx.x & 31;
  const int r    = lane & 15;
  const int hi   = lane >> 4;
  const int m0   = blockIdx.y * 128;
  const int n0   = blockIdx.x * 64;

  const float* qg = Q + ((size_t)b * 1024 + m0) * 128;
  const float* pg = P + ((size_t)b * 1024 + n0) * 128;
  const float* qs = lds;
  const float* ps = lds + QROWS * LROW;
  const unsigned ldsbase = (unsigned)(size_t)(const void*)lds;

  v8f acc[4];
#pragma unroll
  for (int j = 0; j < 4; ++j) acc[j] = (v8f){0.f, 0.f, 0.f, 0.f, 0.f, 0.f, 0.f, 0.f};

  const float* arow = qs + (wave * 16 + r) * LROW;
  const float* brow = ps + r * LROW;

  for (int half = 0; half < 2; ++half) {
    if (wave == 0) {
      tdm_load_tile(ldsbase, qg + half * 64, 64u, (unsigned)QROWS, 128u);
      tdm_load_tile(ldsbase + QROWS * LROW * 4, pg + half * 64, 64u, (unsigned)PROWS, 128u);
      __builtin_amdgcn_s_wait_tensorcnt(0);
    }
    __syncthreads();

#pragma unroll 4
    for (int k = 0; k < 64; k += 4) {
      const int ko = k + (hi << 1);
      v2f a = *(const v2f*)(arow + ko);
#pragma unroll
      for (int j = 0; j < 4; ++j) {
        v2f bb = *(const v2f*)(brow + j * 16 * LROW + ko);
        acc[j] = __builtin_amdgcn_wmma_f32_16x16x4_f32(
            false, a, false, bb, (short)0, acc[j], false, false);
      }
    }
    __syncthreads();
  }

  float* out = sim + (size_t)b * 1024 * 1024;
#pragma unroll
  for (int j = 0; j < 4; ++j) {
#pragma unroll
    for (int i = 0; i < 8; ++i) {
      const int row = m0 + wave * 16 + i + (hi << 3);
      const int col = n0 + j * 16 + r;
      out[(size_t)row * 1024 + col] = acc[j][i];
    }
  }
}

// ---------------------------------------------------------------------------
// Online softmax-denominator combine
// ---------------------------------------------------------------------------
__device__ __forceinline__ void online_comb(float& m, float& sum, float m2, float s2) {
  float nm = fmaxf(m, m2);
  sum = sum * expf(m - nm) + s2 * expf(m2 - nm);
  m = nm;
}

// ---------------------------------------------------------------------------
// K2: column stats (reduce over l for each (b,s)):
//   colmax/colsum = online softmax denom of sim/T over axis 1
//   s1 = (max - mean)/std(ddof=1) of (1+sim)/2 over axis 1
// ---------------------------------------------------------------------------
__global__ __launch_bounds__(256) void k_col_stats(const float* __restrict__ sim,
                                                   float* __restrict__ colmax,
                                                   float* __restrict__ colsum,
                                                   float* __restrict__ s1) {
  __shared__ float sm[4][64], ssum[4][64], sym[4][64], sys[4][64], syq[4][64];
  const int b     = blockIdx.y;
  const int scol  = threadIdx.x & 63;
  const int strip = threadIdx.x >> 6;
  const int s     = blockIdx.x * 64 + scol;
  const float* base = sim + (size_t)b * 1024 * 1024 + s;

  float m = NEG_BIG, sum = 0.f;
  float ym = NEG_BIG, ysum = 0.f, ysq = 0.f;
  for (int l = strip; l < 1024; l += 4) {
    float x = base[(size_t)l * 1024];
    online_comb(m, sum, x * INV_T, 1.0f);
    float y = 0.5f + 0.5f * x;
    ym = fmaxf(ym, y);
    ysum += y;
    ysq += y * y;
  }
  sm[strip][scol] = m;  ssum[strip][scol] = sum;
  sym[strip][scol] = ym; sys[strip][scol] = ysum; syq[strip][scol] = ysq;
  __syncthreads();
  if (strip == 0) {
#pragma unroll
    for (int t = 1; t < 4; ++t) {
      online_comb(m, sum, sm[t][scol], ssum[t][scol]);
      ym = fmaxf(ym, sym[t][scol]);
      ysum += sys[t][scol];
      ysq  += syq[t][scol];
    }
    float mean = ysum * (1.0f / 1024.0f);
    float var  = (ysq - 1024.0f * mean * mean) * (1.0f / 1023.0f);
    float sd   = sqrtf(fmaxf(var, 0.0f));
    int o = b * 1024 + s;
    colmax[o] = m;
    colsum[o] = sum;
    s1[o] = (ym - mean) / sd;
  }
}

// ---------------------------------------------------------------------------
// K3: row stats (reduce over s for each (b,l)), one wave per row (coalesced)
// ---------------------------------------------------------------------------
__global__ __launch_bounds__(256) void k_row_stats(const float* __restrict__ sim,
                                                   float* __restrict__ rowmax,
                                                   float* __restrict__ rowsum,
                                                   float* __restrict__ s2arr) {
  const int b    = blockIdx.y;
  const int wave = threadIdx.x >> 5;
  const int lane = threadIdx.x & 31;
  const int l    = blockIdx.x * 8 + wave;
  const float* row = sim + ((size_t)b * 1024 + l) * 1024;

  float m = NEG_BIG, sum = 0.f, ym = NEG_BIG, ysum = 0.f, ysq = 0.f;
  for (int s = lane; s < 1024; s += 32) {
    float x = row[s];
    online_comb(m, sum, x * INV_T, 1.0f);
    float y = 0.5f + 0.5f * x;
    ym = fmaxf(ym, y);
    ysum += y;
    ysq += y * y;
  }
#pragma unroll
  for (int off = 16; off > 0; off >>= 1) {
    float m2  = __shfl_xor(m, off);
    float s2v = __shfl_xor(sum, off);
    online_comb(m, sum, m2, s2v);
    ym   = fmaxf(ym, __shfl_xor(ym, off));
    ysum += __shfl_xor(ysum, off);
    ysq  += __shfl_xor(ysq, off);
  }
  if (lane == 0) {
    float mean = ysum * (1.0f / 1024.0f);
    float var  = (ysq - 1024.0f * mean * mean) * (1.0f / 1023.0f);
    float sd   = sqrtf(fmaxf(var, 0.0f));
    int o = b * 1024 + l;
    rowmax[o] = m;
    rowsum[o] = sum;
    s2arr[o] = (ym - mean) / sd;
  }
}

// ---------------------------------------------------------------------------
// K4: softmax over a row of 1024 (one block per row); optional 2nd output
// ---------------------------------------------------------------------------
__global__ __launch_bounds__(256) void k_softmax1024(const float* __restrict__ in,
                                                     float* __restrict__ out,
                                                     float* __restrict__ out2) {
  __shared__ float red[256];
  const float* row = in + (size_t)blockIdx.x * 1024;
  const int t = threadIdx.x;
  float x0 = row[t], x1 = row[t + 256], x2 = row[t + 512], x3 = row[t + 768];
  float m = fmaxf(fmaxf(x0, x1), fmaxf(x2, x3));
  red[t] = m;
  __syncthreads();
  for (int off = 128; off > 0; off >>= 1) {
    if (t < off) red[t] = fmaxf(red[t], red[t + off]);
    __syncthreads();
  }
  m = red[0];
  __syncthreads();
  float e0 = expf(x0 - m), e1 = expf(x1 - m), e2 = expf(x2 - m), e3 = expf(x3 - m);
  red[t] = e0 + e1 + e2 + e3;
  __syncthreads();
  for (int off = 128; off > 0; off >>= 1) {
    if (t < off) red[t] += red[t + off];
    __syncthreads();
  }
  float inv = 1.0f / red[0];
  float* o = out + (size_t)blockIdx.x * 1024;
  o[t] = e0 * inv; o[t + 256] = e1 * inv; o[t + 512] = e2 * inv; o[t + 768] = e3 * inv;
  if (out2) {
    float* o2 = out2 + (size_t)blockIdx.x * 1024;
    o2[t] = e0 * inv; o2[t + 256] = e1 * inv; o2[t + 512] = e2 * inv; o2[t + 768] = e3 * inv;
  }
}

// ---------------------------------------------------------------------------
// K5: fused loss accumulation over (b,l,s). One wave per row.
// acc[b*8 + {0:psum,1:pcnt,2:nsum,3:ncnt,4:l1,5:l2,6:l3}]
// ---------------------------------------------------------------------------
__global__ __launch_bounds__(256) void k_main_acc(const float* __restrict__ sim,
                                                  const int* __restrict__ label,
                                                  const float* __restrict__ colmax,
                                                  const float* __restrict__ colsum,
                                                  const float* __restrict__ rowmax,
                                                  const float* __restrict__ rowsum,
                                                  const float* __restrict__ sharp1,
                                                  const float* __restrict__ sharp2,
                                                  float* __restrict__ acc) {
  const int b    = blockIdx.y;
  const int wave = threadIdx.x >> 5;
  const int lane = threadIdx.x & 31;
  const int l    = blockIdx.x * 8 + wave;
  const size_t bl = (size_t)b * 1024 + l;
  const float* row  = sim + bl * 1024;
  const int* lrow   = label + bl * 1024;
  const float rm    = rowmax[bl];
  const float rinv  = 1.0f / rowsum[bl];
  const float sh2   = sharp2[bl];
  const float* cm   = colmax + b * 1024;
  const float* cs   = colsum + b * 1024;
  const float* s1p  = sharp1 + b * 1024;

  float psum = 0.f, pcnt = 0.f, nsum = 0.f, ncnt = 0.f, l1 = 0.f, l2 = 0.f, l3 = 0.f;
  for (int s = lane; s < 1024; s += 32) {
    float x = row[s];
    int lab = lrow[s];
    float g = x * INV_T;
    float conf = (expf(g - cm[s]) / cs[s]) * (expf(g - rm) * rinv);
    conf = fminf(fmaxf(conf, EPSV), 1.0f - EPSV);
    float y  = 0.5f + 0.5f * x;
    float yc = fminf(fmaxf(y, EPSV), 1.0f - EPSV);
    if (lab == 1) {
      float nll = -logf(yc);
      psum += -logf(conf);
      pcnt += 1.0f;
      l1 += nll * s1p[s];
      l2 += nll;
    } else if (lab == 0) {
      nsum += -logf(conf);
      ncnt += 1.0f;
      l3 += -logf(1.0f - yc);
    }
  }
#pragma unroll
  for (int off = 16; off > 0; off >>= 1) {
    psum += __shfl_xor(psum, off);
    pcnt += __shfl_xor(pcnt, off);
    nsum += __shfl_xor(nsum, off);
    ncnt += __shfl_xor(ncnt, off);
    l1   += __shfl_xor(l1, off);
    l2   += __shfl_xor(l2, off);
    l3   += __shfl_xor(l3, off);
  }
  if (lane == 0) {
    float* a = acc + b * 8;
    atomicAdd(a + 0, psum);
    atomicAdd(a + 1, pcnt);
    atomicAdd(a + 2, nsum);
    atomicAdd(a + 3, ncnt);
    atomicAdd(a + 4, l1);
    atomicAdd(a + 5, l2 * sh2);
    atomicAdd(a + 6, l3);
  }
}

// ---------------------------------------------------------------------------
// K6a: normalized selected queries + cl_pos_save output
// ---------------------------------------------------------------------------
__global__ __launch_bounds__(128) void k_cl_prep(const float* __restrict__ q_cl,
                                                 const float* __restrict__ cl_pos,
                                                 const int* __restrict__ idx,
                                                 float* __restrict__ qnorm,
                                                 float* __restrict__ out_sel) {
  __shared__ float red[128];
  const int n = blockIdx.x, c = threadIdx.x;
  const int in = idx[n];
  float qv = q_cl[((size_t)n * 256 + in) * 128 + c];
  red[c] = qv * qv;
  __syncthreads();
  for (int off = 64; off > 0; off >>= 1) {
    if (c < off) red[c] += red[c + off];
    __syncthreads();
  }
  float inv = 1.0f / fmaxf(sqrtf(red[0]), 1e-12f);
  qnorm[n * 128 + c]   = qv * inv;
  out_sel[n * 128 + c] = cl_pos[((size_t)n * 256 + in) * 128 + c];
}

// ---------------------------------------------------------------------------
// K6b: contrastive sums. Thread j owns all_k row j (2048 rows).
// ---------------------------------------------------------------------------
__global__ __launch_bounds__(256) void k_contrast(const float* __restrict__ cl_pos,
                                                  const int* __restrict__ idx,
                                                  const float* __restrict__ qnorm,
                                                  float* __restrict__ cacc) {
  __shared__ float qn[8 * 128];
  __shared__ float red[256];
  const int t = threadIdx.x;
#pragma unroll
  for (int i = 0; i < 4; ++i) qn[t + 256 * i] = qnorm[t + 256 * i];
  __syncthreads();

  const int j = blockIdx.x * 256 + t;
  int n, p;
  if (j < 8) {
    n = j;
    p = idx[j];
  } else {
    int u = j - 8;
    n = u / 255;
    p = u % 255 + 1;
    if (p == idx[n]) p = 0;  // this slot got the swapped-in first row
  }
  const float* krow = cl_pos + ((size_t)n * 256 + p) * 128;
  float sq = 0.f;
  for (int c = 0; c < 128; ++c) { float v = krow[c]; sq += v * v; }
  float inv = 1.0f / fmaxf(sqrtf(sq), 1e-12f);

  float d[8];
#pragma unroll
  for (int i = 0; i < 8; ++i) d[i] = 0.f;
  for (int c = 0; c < 128; ++c) {
    float kc = krow[c] * inv;
#pragma unroll
    for (int i = 0; i < 8; ++i) d[i] += qn[i * 128 + c] * kc;
  }

  float pos = 0.f, neg = 0.f;
  if (j < 8) {
    float dj = 0.f;
#pragma unroll
    for (int i = 0; i < 8; ++i) if (i == j) dj = d[i];
    float sgl = fminf(fmaxf(0.5f + 0.5f * dj, EPSV), 1.0f - EPSV);
    pos = -logf(sgl);
  } else {
#pragma unroll
    for (int i = 0; i < 8; ++i) {
      float sgl = fminf(fmaxf(0.5f + 0.5f * d[i], EPSV), 1.0f - EPSV);
      neg += -logf(1.0f - sgl);
    }
  }
  red[t] = pos;
  __syncthreads();
  for (int off = 128; off > 0; off >>= 1) {
    if (t < off) red[t] += red[t + off];
    __syncthreads();
  }
  if (t == 0) atomicAdd(cacc + 0, red[0]);
  __syncthreads();
  red[t] = neg;
  __syncthreads();
  for (int off = 128; off > 0; off >>= 1) {
    if (t < off) red[t] += red[t + off];
    __syncthreads();
  }
  if (t == 0) atomicAdd(cacc + 1, red[0]);
}

// ---------------------------------------------------------------------------
// K7: final scalars -> out[9216..9220] = total, geo_pos, geo_neg, sharp, cl
// ---------------------------------------------------------------------------
__global__ void k_finalize(const float* __restrict__ acc,
                           const float* __restrict__ cacc,
                           float* __restrict__ o) {
  if (threadIdx.x != 0 || blockIdx.x != 0) return;
  float gp = 0.f, gn = 0.f, sh = 0.f;
  for (int b = 0; b < 8; ++b) {
    const float* a = acc + b * 8;
    float pm  = a[0] / fmaxf(a[1], 1.0f);
    float nm  = a[2] / fmaxf(a[3], 1.0f);
    float l3m = a[6] / fmaxf(a[3], 1.0f);
    gp += pm;
    gn += nm;
    sh += 0.5f * (a[4] + a[5]) + l3m;
  }
  gp *= 0.125f;
  gn *= 0.125f;
  sh *= 0.125f;
  float loss_geo   = gp + gn;
  float loss_geo_w = (0.5f * loss_geo + 0.5f * sh) * 0.5f;   // W_SHARP, W_LOCAL
  float loss_cl = ((cacc[0] * (1.0f / 8.0f)) + (cacc[1] * (1.0f / 16320.0f))) * 0.5f * 0.5f;
  o[0] = loss_geo_w + loss_cl;
  o[1] = gp;
  o[2] = gn;
  o[3] = sh;
  o[4] = loss_cl;
}

// ---------------------------------------------------------------------------
extern "C" void kernel_launch(void* const* d_in, const int* in_sizes, int n_in,
                              void* d_out, int out_size, void* d_ws, size_t ws_size,
                              hipStream_t stream) {
  (void)in_sizes; (void)n_in; (void)out_size; (void)ws_size;
  const float* q_geo   = (const float*)d_in[0];
  const float* q_cl    = (const float*)d_in[1];
  const float* geo_pos = (const float*)d_in[2];
  const float* cl_pos  = (const float*)d_in[3];
  const int*   label   = (const int*)d_in[4];
  const int*   idx     = (const int*)d_in[5];
  float* out = (float*)d_out;
  float* ws  = (float*)d_ws;

  // workspace layout (floats)
  float* sim    = ws;                    // 8*1024*1024
  float* colmax = ws + 8388608;          // 8192
  float* colsum = colmax + 8192;
  float* s1     = colsum + 8192;
  float* rowmax = s1 + 8192;
  float* rowsum = rowmax + 8192;
  float* s2     = rowsum + 8192;
  float* sharp1 = s2 + 8192;
  float* sharp2 = sharp1 + 8192;
  float* qnorm  = sharp2 + 8192;         // 1024
  float* acc    = qnorm + 1024;          // 64 (8 batches x 8 slots)
  float* cacc   = acc + 64;              // 2

  k_zero<<<1, 128, 0, stream>>>(acc, 66);
  k_sim_gemm<<<dim3(16, 8, 8), 256, 0, stream>>>(q_geo, geo_pos, sim);
  k_col_stats<<<dim3(16, 8), 256, 0, stream>>>(sim, colmax, colsum, s1);
  k_row_stats<<<dim3(128, 8), 256, 0, stream>>>(sim, rowmax, rowsum, s2);
  k_softmax1024<<<8, 256, 0, stream>>>(s1, sharp1, out);          // sharpness output
  k_softmax1024<<<8, 256, 0, stream>>>(s2, sharp2, (float*)nullptr);
  k_main_acc<<<dim3(128, 8), 256, 0, stream>>>(sim, label, colmax, colsum,
                                               rowmax, rowsum, sharp1, sharp2, acc);
  k_cl_prep<<<8, 128, 0, stream>>>(q_cl, cl_pos, idx, qnorm, out + 8192);
  k_contrast<<<8, 256, 0, stream>>>(cl_pos, idx, qnorm, cacc);
  k_finalize<<<1, 32, 0, stream>>>(acc, cacc, out + 9216);
}